// Convolution_24970939858998
// MI455X (gfx1250) — compile-verified
//
#include <hip/hip_runtime.h>

typedef __attribute__((ext_vector_type(2))) float v2f;
typedef __attribute__((ext_vector_type(8))) float v8f;

namespace {
constexpr int BATCH = 32;
constexpr int CIN   = 8;
constexpr int COUT  = 8;
constexpr int HDIM  = 1024;
constexpr int WDIM  = 1024;
constexpr int TILE_H = 16;             // output rows per block
constexpr int TILE_W = 64;             // output cols per block
constexpr int LDS_ROWS = TILE_H + 2;   // 18 (halo)
constexpr int LDS_COLS = TILE_W + 2;   // 66 (halo)
constexpr int PAIR_ELEMS = LDS_ROWS * LDS_COLS;  // 1188 v2f per channel-pair
// pair stride padded so 2*PAIR_STRIDE (words) % 64 == 32:
// the two wave halves (pair +0 vs +1) hit disjoint LDS bank sets on b64 reads.
constexpr int PAIR_STRIDE = 1200;      // v2f units; 2400 words, 2400%64==32
constexpr int NPAIR = CIN / 2;         // 4 channel pairs
constexpr int KTOT  = 96;              // 8 cin * 4 input rows * 3 kw
constexpr int NSTEP = KTOT / 4;        // 24 wmma f32_16x16x4 per strip
}

__global__ __launch_bounds__(256)
void conv3x3_wmma_f32(const float* __restrict__ in,
                      const float* __restrict__ wgt,
                      const float* __restrict__ bias,
                      float* __restrict__ out) {
  // channel-pair interleaved input tile: lds2[pair*1200 + row*66 + col] = {c=2p, c=2p+1}
  __shared__ v2f lds2[NPAIR * PAIR_STRIDE];   // 38400 B

  const int tid  = threadIdx.x;
  const int lane = tid & 31;
  const int wv   = tid >> 5;        // 8 waves
  const int hi   = lane >> 4;       // wave half
  const int nm   = lane & 15;       // N index (pixel) for B ; M index for A
  const int koff = hi * 2;          // K sub-pair owned by this half (A/B layout)

  const int b  = blockIdx.z;
  const int h0 = blockIdx.y * TILE_H;
  const int w0 = blockIdx.x * TILE_W;

  // ---- cooperative stage of haloed input tile into LDS (zero padded) ----
  // one v2f (two channels, same pixel) per element -> single ds_store_b64
  for (int i = tid; i < NPAIR * PAIR_ELEMS; i += 256) {
    int pair = i / PAIR_ELEMS;
    int rem  = i - pair * PAIR_ELEMS;
    int r    = rem / LDS_COLS;
    int col  = rem - r * LDS_COLS;
    int gr = h0 - 1 + r;
    int gc = w0 - 1 + col;
    v2f v = {0.0f, 0.0f};
    if ((unsigned)gr < (unsigned)HDIM && (unsigned)gc < (unsigned)WDIM) {
      int gidx = ((b * CIN + 2 * pair) * HDIM + gr) * WDIM + gc;
      v.x = in[gidx];
      v.y = in[gidx + HDIM * WDIM];
    }
    lds2[pair * PAIR_STRIDE + rem] = v;
  }

  // ---- A fragments: weights, K-order k = kh4*24 + kw*8 + c ----
  // M = 16 = cout(8) x two row offsets. m<8: row h (kh = kh4, valid kh4<3)
  //                                     m>=8: row h+1 (kh = kh4-1, valid kh4>=1)
  // A 16x4 layout: lanes 0-15 hold M=lane; .x = K=4t+koff, .y = K=4t+koff+1.
  const bool lowM = (nm < COUT);
  const int  cw   = lowM ? nm : (nm - COUT);   // cout for this M row
  v2f aw[NSTEP];
#pragma unroll
  for (int t = 0; t < NSTEP; ++t) {
    const int g   = t >> 1;           // (4t)/8 = kh4*3 + kw (compile-time)
    const int kh4 = g / 3;
    const int kw  = g - kh4 * 3;
    const int cb  = (4 * t) & 7;      // channel base (0 or 4)
    const int c0  = cb + koff;
    float ax = 0.0f, ay = 0.0f;
    if (lowM) {
      if (kh4 < 3) {
        ax = wgt[cw * 72 + c0 * 9 + kh4 * 3 + kw];
        ay = wgt[cw * 72 + (c0 + 1) * 9 + kh4 * 3 + kw];
      }
    } else {
      if (kh4 >= 1) {
        ax = wgt[cw * 72 + c0 * 9 + (kh4 - 1) * 3 + kw];
        ay = wgt[cw * 72 + (c0 + 1) * 9 + (kh4 - 1) * 3 + kw];
      }
    }
    aw[t].x = ax;
    aw[t].y = ay;
  }

  // bias per cout (uniform scalars -> s_loads)
  float bs[COUT];
#pragma unroll
  for (int r = 0; r < COUT; ++r) bs[r] = bias[r];

  __syncthreads();

  // ---- each wave: row pair (2*wv, 2*wv+1), 4 column segments of 16 ----
  const int rp = wv;
  for (int cs = 0; cs < 4; ++cs) {
    const int s = cs * 16;
    // B fragment (one v2f = channels c0,c0+1 of one pixel) per step:
    //   lds2[(cb/2 + hi)*1200 + (2rp + kh4)*66 + (s + nm + kw)]
    const int base = hi * PAIR_STRIDE + (2 * rp) * LDS_COLS + s + nm;

    v8f acc = {};                          // bias added at store
#pragma unroll
    for (int t = 0; t < NSTEP; ++t) {
      const int g   = t >> 1;
      const int kh4 = g / 3;
      const int kw  = g - kh4 * 3;
      const int cb  = (4 * t) & 7;
      const int off = (cb >> 1) * PAIR_STRIDE + kh4 * LDS_COLS + kw;  // compile-time
      v2f bf = lds2[base + off];           // single ds_load_b64, even VGPR pair
      // D = A(16x4 f32) * B(4x16 f32) + C(16x16 f32)
      acc = __builtin_amdgcn_wmma_f32_16x16x4_f32(
          /*neg_a=*/false, aw[t], /*neg_b=*/false, bf,
          /*c_mod=*/(short)0, acc, /*reuse_a=*/false, /*reuse_b=*/false);
    }

    // C layout: VGPR r -> lanes 0-15: M=r (cout r, row h), lanes 16-31: M=r+8
    // (cout r, row h+1). All 32 lanes store; 2x 16 consecutive floats per cout.
    const int gh = h0 + 2 * rp + hi;       // output row for this lane half
    const int gc = w0 + s + nm;            // output col
    const int obase = ((b * COUT) * HDIM + gh) * WDIM + gc;
#pragma unroll
    for (int r = 0; r < COUT; ++r)
      __builtin_nontemporal_store(acc[r] + bs[r], &out[obase + r * (HDIM * WDIM)]);
  }
}

extern "C" void kernel_launch(void* const* d_in, const int* in_sizes, int n_in,
                              void* d_out, int out_size, void* d_ws, size_t ws_size,
                              hipStream_t stream) {
  (void)in_sizes; (void)n_in; (void)d_ws; (void)ws_size; (void)out_size;
  const float* x    = (const float*)d_in[0];
  const float* wgt  = (const float*)d_in[1];
  const float* bias = (const float*)d_in[2];
  float* out        = (float*)d_out;

  dim3 grid(WDIM / TILE_W, HDIM / TILE_H, BATCH);  // (16, 64, 32)
  conv3x3_wmma_f32<<<grid, 256, 0, stream>>>(x, wgt, bias, out);
}